// GaussianProbeCompressor3D_18734647345464
// MI455X (gfx1250) — compile-verified
//
#include <hip/hip_runtime.h>
#include <math.h>

// Sizes fixed by the reference
#define KG 256   // gaussians
#define DF 12    // latent dim
#define HN 32    // hidden
#define CN 9     // sh channels
#define EPSF 1e-8f

// padded LDS row strides (odd -> conflict-free for every access pattern used)
#define WS 17    // weights tile stride   ([k][m], m<16)
#define MS 17    // mlp_in tile stride    ([M][col], col<16)
#define HS 33    // hidden tile stride    ([M][col], col<32)

typedef __attribute__((ext_vector_type(2))) float f32x2;
typedef __attribute__((ext_vector_type(8))) float f32x8;

__device__ __forceinline__ f32x8 wmma4(f32x2 a, f32x2 b, f32x8 c) {
  // V_WMMA_F32_16X16X4_F32 : D = A(16x4) * B(4x16) + C(16x16)
  return __builtin_amdgcn_wmma_f32_16x16x4_f32(false, a, false, b, (short)0, c,
                                               false, false);
}

// ---------------------------------------------------------------------------
// Per-gaussian precompute: mu(3), precision (diag + pre-doubled off-diag, 6),
// radius^2 (1) -> 12 f32 per gaussian.
// ---------------------------------------------------------------------------
__global__ void gk_precompute(const float* __restrict__ mu,
                              const float* __restrict__ log_s,
                              const float* __restrict__ q,
                              float* __restrict__ gk) {
  int k = blockIdx.x * blockDim.x + threadIdx.x;
  if (k >= KG) return;
  float qw = q[k * 4 + 0], qx = q[k * 4 + 1], qy = q[k * 4 + 2], qz = q[k * 4 + 3];
  float inv = 1.0f / (sqrtf(qw * qw + qx * qx + qy * qy + qz * qz) + EPSF);
  qw *= inv; qx *= inv; qy *= inv; qz *= inv;
  float r00 = 1.f - 2.f * (qy * qy + qz * qz);
  float r01 = 2.f * (qx * qy - qw * qz);
  float r02 = 2.f * (qx * qz + qw * qy);
  float r10 = 2.f * (qx * qy + qw * qz);
  float r11 = 1.f - 2.f * (qx * qx + qz * qz);
  float r12 = 2.f * (qy * qz - qw * qx);
  float r20 = 2.f * (qx * qz - qw * qy);
  float r21 = 2.f * (qy * qz + qw * qx);
  float r22 = 1.f - 2.f * (qx * qx + qy * qy);
  float s0 = expf(log_s[k * 3 + 0]);
  float s1 = expf(log_s[k * 3 + 1]);
  float s2 = expf(log_s[k * 3 + 2]);
  float i0 = 1.f / (s0 * s0 + EPSF);
  float i1 = 1.f / (s1 * s1 + EPSF);
  float i2 = 1.f / (s2 * s2 + EPSF);
  // P = R diag(i) R^T (symmetric); off-diagonals stored pre-doubled
  float p00 = r00 * r00 * i0 + r01 * r01 * i1 + r02 * r02 * i2;
  float p01 = r00 * r10 * i0 + r01 * r11 * i1 + r02 * r12 * i2;
  float p02 = r00 * r20 * i0 + r01 * r21 * i1 + r02 * r22 * i2;
  float p11 = r10 * r10 * i0 + r11 * r11 * i1 + r12 * r12 * i2;
  float p12 = r10 * r20 * i0 + r11 * r21 * i1 + r12 * r22 * i2;
  float p22 = r20 * r20 * i0 + r21 * r21 * i1 + r22 * r22 * i2;
  float rad = 3.f * fmaxf(s0, fmaxf(s1, s2));
  float* g = gk + (size_t)k * 12;
  g[0] = mu[k * 3 + 0]; g[1] = mu[k * 3 + 1]; g[2] = mu[k * 3 + 2];
  g[3] = p00; g[4] = 2.f * p01; g[5] = 2.f * p02;
  g[6] = p11; g[7] = 2.f * p12; g[8] = p22;
  g[9] = rad * rad; g[10] = 0.f; g[11] = 0.f;
}

// ---------------------------------------------------------------------------
// Main kernel: 4 waves/block, each wave owns a 16-probe tile.
// ---------------------------------------------------------------------------
__global__ __launch_bounds__(128)
void probe_main(const float* __restrict__ probe,
                const float* __restrict__ F,
                const float* __restrict__ W1, const float* __restrict__ b1,
                const float* __restrict__ W2, const float* __restrict__ b2,
                const float* __restrict__ gk,
                float* __restrict__ out_sh, float* __restrict__ out_lat,
                float* __restrict__ out_w, int N) {
  __shared__ float s_gk[KG * 12];          // mu, precision, r^2
  __shared__ float s_F[KG * 16];           // F padded 12->16 cols
  __shared__ float s_W1[16 * HN];          // W1 padded 15->16 rows
  __shared__ float s_W2[HN * 16];          // W2 padded 9->16 cols
  __shared__ float s_b1[HN];
  __shared__ float s_b2[16];
  __shared__ float s_w[4][KG * WS];        // [wave][k][m], stride 17
  __shared__ float s_mlp[4][16 * MS];      // [wave][M][col], stride 17
  __shared__ float s_h[4][16 * HS];        // [wave][M][col], stride 33
  __shared__ float s_inv[4][16];

  const int tid = threadIdx.x;
  for (int i = tid; i < KG * 12; i += 128) s_gk[i] = gk[i];
  for (int i = tid; i < KG * 16; i += 128) {
    int k = i >> 4, c = i & 15;
    s_F[i] = (c < DF) ? F[k * DF + c] : 0.f;
  }
  for (int i = tid; i < 16 * HN; i += 128) {
    int r = i >> 5;
    s_W1[i] = (r < DF + 3) ? W1[i] : 0.f;   // W1 is (15 x 32) row-major
  }
  for (int i = tid; i < HN * 16; i += 128) {
    int r = i >> 4, c = i & 15;
    s_W2[i] = (c < CN) ? W2[r * CN + c] : 0.f;
  }
  if (tid < HN) s_b1[tid] = b1[tid];
  if (tid < 16) s_b2[tid] = (tid < CN) ? b2[tid] : 0.f;
  __syncthreads();

  const int wave = tid >> 5;
  const int lane = tid & 31;
  const int m  = lane & 15;     // probe row within tile (A/M) and column (B,D/N)
  const int hh = lane >> 4;     // lane-half: K pair {2hh, 2hh+1} of each 4-step in WMMA fragments
  const int base = (blockIdx.x * 4 + wave) * 16;
  const int row = base + m;
  const int rc = (row < N) ? row : (N > 0 ? N - 1 : 0);
  const float px = probe[rc * 3 + 0];
  const float py = probe[rc * 3 + 1];
  const float pz = probe[rc * 3 + 2];

  // ---- pass 1: gaussian responsibilities for this wave's 16 probes ----
  float* wrow = s_w[wave];
  float acc = 0.f;
#pragma unroll 4
  for (int j = 0; j < KG / 2; ++j) {
    const int k = 2 * j + hh;                 // half-uniform -> LDS broadcast
    const float* g = &s_gk[k * 12];
    const float dx = px - g[0], dy = py - g[1], dz = pz - g[2];
    // pure FMA chain: off-diagonals pre-doubled
    const float mah = g[3] * dx * dx + g[6] * dy * dy + g[8] * dz * dz +
                      g[4] * (dx * dy) + g[5] * (dx * dz) + g[7] * (dy * dz);
    const float d2 = dx * dx + dy * dy + dz * dz;
    const float gv = (d2 < g[9]) ? __expf(-0.5f * mah) : 0.f;
    wrow[k * WS + m] = gv;                    // conflict-free (stride 17)
    acc += gv;
  }
  const float tot = acc + __shfl_xor(acc, 16, 32);
  const float winv = 1.f / (tot + EPSF);
  if (hh == 0) s_inv[wave][m] = winv;
  __syncthreads();

  // ---- normalized weights -> global (coalesced; stride-17 LDS reads hit
  //      32 distinct banks for 32 consecutive k) ----
  for (int idx = lane; idx < 16 * KG; idx += 32) {
    const int mm = idx >> 8;
    const int k = idx & (KG - 1);
    const int rr = base + mm;
    if (rr < N) out_w[(size_t)rr * KG + k] = wrow[k * WS + mm] * s_inv[wave][mm];
  }

  // ---- latent = weights @ F  (16x16x4 f32 WMMA, 64 K-steps) ----
  f32x8 cl = {};
#pragma unroll 8
  for (int k0 = 0; k0 < KG; k0 += 4) {
    const int kk = k0 + 2 * hh;
    f32x2 a, b;
    a.x = wrow[kk * WS + m] * winv;
    a.y = wrow[(kk + 1) * WS + m] * winv;
    b.x = s_F[kk * 16 + m];
    b.y = s_F[(kk + 1) * 16 + m];
    cl = wmma4(a, b, cl);
  }

  // ---- write latent, stage mlp_in = [latent | pos | 0] ----
  float* mrow = s_mlp[wave];
#pragma unroll
  for (int r = 0; r < 8; ++r) {
    const int M = r + 8 * hh;
    mrow[M * MS + m] = cl[r];
    const int rr = base + M;
    if (m < DF && rr < N) out_lat[(size_t)rr * DF + m] = cl[r];
  }
  if (hh == 0) {
    mrow[m * MS + 12] = px;
    mrow[m * MS + 13] = py;
    mrow[m * MS + 14] = pz;
    mrow[m * MS + 15] = 0.f;
  }
  __syncthreads();

  // ---- layer 1: h = relu(mlp_in @ W1 + b1), N=32 -> two D tiles ----
  f32x8 h0 = {}, h1 = {};
#pragma unroll
  for (int k0 = 0; k0 < 16; k0 += 4) {
    const int kk = k0 + 2 * hh;
    f32x2 a, b0, b1v;
    a.x = mrow[m * MS + kk];
    a.y = mrow[m * MS + kk + 1];
    b0.x = s_W1[kk * HN + m];
    b0.y = s_W1[(kk + 1) * HN + m];
    b1v.x = s_W1[kk * HN + 16 + m];
    b1v.y = s_W1[(kk + 1) * HN + 16 + m];
    h0 = wmma4(a, b0, h0);
    h1 = wmma4(a, b1v, h1);
  }
  float* hrow = s_h[wave];
#pragma unroll
  for (int r = 0; r < 8; ++r) {
    const int M = r + 8 * hh;
    hrow[M * HS + m]      = fmaxf(h0[r] + s_b1[m], 0.f);
    hrow[M * HS + 16 + m] = fmaxf(h1[r] + s_b1[16 + m], 0.f);
  }
  __syncthreads();

  // ---- layer 2: sh = h @ W2 + b2 ----
  f32x8 o = {};
#pragma unroll
  for (int k0 = 0; k0 < HN; k0 += 4) {
    const int kk = k0 + 2 * hh;
    f32x2 a, b;
    a.x = hrow[m * HS + kk];
    a.y = hrow[m * HS + kk + 1];
    b.x = s_W2[kk * 16 + m];
    b.y = s_W2[(kk + 1) * 16 + m];
    o = wmma4(a, b, o);
  }
#pragma unroll
  for (int r = 0; r < 8; ++r) {
    const int M = r + 8 * hh;
    const int rr = base + M;
    if (m < CN && rr < N) out_sh[(size_t)rr * CN + m] = o[r] + s_b2[m];
  }
}

// ---------------------------------------------------------------------------
extern "C" void kernel_launch(void* const* d_in, const int* in_sizes, int n_in,
                              void* d_out, int out_size, void* d_ws, size_t ws_size,
                              hipStream_t stream) {
  (void)n_in; (void)out_size; (void)ws_size;
  const float* probe = (const float*)d_in[0];
  const float* mu    = (const float*)d_in[1];
  const float* log_s = (const float*)d_in[2];
  const float* q     = (const float*)d_in[3];
  const float* F     = (const float*)d_in[4];
  const float* W1    = (const float*)d_in[5];
  const float* b1    = (const float*)d_in[6];
  const float* W2    = (const float*)d_in[7];
  const float* b2    = (const float*)d_in[8];
  const int N = in_sizes[0] / 3;

  float* gk = (float*)d_ws;  // KG * 12 floats
  float* out_sh  = (float*)d_out;
  float* out_lat = out_sh + (size_t)N * CN;
  float* out_w   = out_lat + (size_t)N * DF;

  gk_precompute<<<1, KG, 0, stream>>>(mu, log_s, q, gk);
  const int blocks = (N + 63) / 64;  // 64 probes per block (4 waves x 16)
  probe_main<<<blocks, 128, 0, stream>>>(probe, F, W1, b1, W2, b2, gk,
                                         out_sh, out_lat, out_w, N);
}